// SetConvLayer_76639396430011
// MI455X (gfx1250) — compile-verified
//
#include <hip/hip_runtime.h>
#include <math.h>

typedef float v2f __attribute__((ext_vector_type(2)));
typedef float v8f __attribute__((ext_vector_type(8)));

#define NSAMP   2000000
#define NBLK    1024
#define WPB     8                       // waves per block (256 threads, wave32)
#define TOTAL_WAVES (NBLK * WPB)        // 8192
#define NGROUP  (NSAMP / 4)             // 500000 K-groups of 4 samples
#define GPW     ((NGROUP + TOTAL_WAVES - 1) / TOTAL_WAVES)   // 62 groups/wave
#define EPART   1056                    // 1024 G entries + 32 s entries
#define NSLOT   32

// One K-group (4 samples) of loads + 4 WMMAs into the given accumulator set.
// A = feature^T tile (M = feature column), B = x tile (N = x column), K = n.
__device__ __forceinline__ void kstep(const float* __restrict__ X,
                                      const float* __restrict__ F,
                                      int base,
                                      v8f& g00, v8f& g01, v8f& g10, v8f& g11,
                                      float& s0, float& s1) {
    v2f a0, a1, b0, b1;
    a0.x = __builtin_nontemporal_load(F + base);        // K=2h,   col c
    a0.y = __builtin_nontemporal_load(F + base + 32);   // K=2h+1, col c
    a1.x = __builtin_nontemporal_load(F + base + 16);   // col c+16
    a1.y = __builtin_nontemporal_load(F + base + 48);
    b0.x = __builtin_nontemporal_load(X + base);
    b0.y = __builtin_nontemporal_load(X + base + 32);
    b1.x = __builtin_nontemporal_load(X + base + 16);
    b1.y = __builtin_nontemporal_load(X + base + 48);

    g00 = __builtin_amdgcn_wmma_f32_16x16x4_f32(false, a0, false, b0,
                                                (short)0, g00, false, false);
    g01 = __builtin_amdgcn_wmma_f32_16x16x4_f32(false, a0, false, b1,
                                                (short)0, g01, false, false);
    g10 = __builtin_amdgcn_wmma_f32_16x16x4_f32(false, a1, false, b0,
                                                (short)0, g10, false, false);
    g11 = __builtin_amdgcn_wmma_f32_16x16x4_f32(false, a1, false, b1,
                                                (short)0, g11, false, false);

    s0 += a0.x + a0.y;      // column c       (both half-waves duplicate cols)
    s1 += a1.x + a1.y;      // column c+16
}

// ---------------------------------------------------------------------------
// Stage 1: G[i,j] = sum_n feature[n,i]*x[n,j]  (32x32, fp32 WMMA 16x16x4)
//          s[i]   = sum_n feature[n,i]
// ---------------------------------------------------------------------------
__global__ __launch_bounds__(256) void g_stage1(const float* __restrict__ X,
                                                const float* __restrict__ F,
                                                float* __restrict__ partial) {
    const int tid  = threadIdx.x;
    const int lane = tid & 31;
    const int h    = lane >> 4;     // half-wave select
    const int c    = lane & 15;     // column within tile

    // Scalarize the wave id so the trip count lives in SGPRs (s_cbranch loop,
    // no per-iteration exec masking).
    const int wid = __builtin_amdgcn_readfirstlane(blockIdx.x * WPB + (tid >> 5));

    // Two accumulator sets: consecutive WMMAs on the same tile are independent,
    // breaking the serial D->C latency chain.
    v8f g00a = {}, g01a = {}, g10a = {}, g11a = {};
    v8f g00b = {}, g01b = {}, g10b = {}, g11b = {};
    float s0 = 0.0f, s1 = 0.0f;

    int gBeg = wid * GPW;
    int gEnd = gBeg + GPW;
    if (gEnd > NGROUP) gEnd = NGROUP;
    if (gBeg > NGROUP) gBeg = NGROUP;

    const int lofs = (h << 1) * 32 + c;        // per-lane offset within a group

    int g = gBeg;
    // Main loop: 4 K-groups (16 samples) per iteration -> 32 NT loads in
    // flight per wave, 16 WMMAs alternating between the two accumulator sets.
    for (; g + 4 <= gEnd; g += 4) {
        const int b0 = (g << 2) * 32 + lofs;
        kstep(X, F, b0,       g00a, g01a, g10a, g11a, s0, s1);
        kstep(X, F, b0 + 128, g00b, g01b, g10b, g11b, s0, s1);
        kstep(X, F, b0 + 256, g00a, g01a, g10a, g11a, s0, s1);
        kstep(X, F, b0 + 384, g00b, g01b, g10b, g11b, s0, s1);
    }
    for (; g < gEnd; ++g) {
        kstep(X, F, (g << 2) * 32 + lofs, g00a, g01a, g10a, g11a, s0, s1);
    }

    // --- deterministic in-block combine via per-wave LDS slices -----------
    __shared__ float Gs[WPB * 1024];
    __shared__ float Ss[WPB * 64];

    const int wv = tid >> 5;
    float* gw = Gs + wv * 1024;
    // C/D layout: VGPR p -> row p (lanes 0-15) / row p+8 (lanes 16-31), col c
    #pragma unroll
    for (int p = 0; p < 8; ++p) {
        const int r = h ? (p + 8) : p;
        gw[r * 32 + c]             = g00a[p] + g00b[p];
        gw[r * 32 + 16 + c]        = g01a[p] + g01b[p];
        gw[(16 + r) * 32 + c]      = g10a[p] + g10b[p];
        gw[(16 + r) * 32 + 16 + c] = g11a[p] + g11b[p];
    }
    Ss[wv * 64 + lane]      = s0;   // lanes l and l+16 both carry col (l&15)
    Ss[wv * 64 + 32 + lane] = s1;
    __syncthreads();

    float* outp = partial + (size_t)blockIdx.x * EPART;
    for (int e = tid; e < EPART; e += 256) {
        float acc = 0.0f;
        if (e < 1024) {
            #pragma unroll
            for (int w = 0; w < WPB; ++w) acc += Gs[w * 1024 + e];
        } else {
            const int cc  = e - 1024;
            const int off = (cc < 16) ? 0 : 32;   // s0 region vs s1 region
            const int cl  = cc & 15;
            #pragma unroll
            for (int w = 0; w < WPB; ++w)
                acc += Ss[w * 64 + off + cl] + Ss[w * 64 + off + 16 + cl];
        }
        outp[e] = acc;
    }
}

// ---------------------------------------------------------------------------
// Stage 2a: fixed-order tree reduction of 1024 block partials -> 32 slots
// ---------------------------------------------------------------------------
__global__ void g_stage2a(const float* __restrict__ partial,
                          float* __restrict__ slotred) {
    const int idx = blockIdx.x * 256 + threadIdx.x;
    if (idx >= NSLOT * EPART) return;
    const int k = idx / EPART;
    const int e = idx - k * EPART;
    float acc = 0.0f;
    #pragma unroll 4
    for (int m = 0; m < NBLK / NSLOT; ++m)
        acc += partial[(size_t)(k + NSLOT * m) * EPART + e];
    slotred[idx] = acc;
}

// ---------------------------------------------------------------------------
// Stage 2b: reduce 32 slots -> G,s ; t = G*fcW^T + bias x s ;
//           out[o] = (1/N) sum_i softmax(w)[i,o] * t[i,o]
// ---------------------------------------------------------------------------
__global__ __launch_bounds__(256) void g_stage2b(const float* __restrict__ slotred,
                                                 const float* __restrict__ fcW,
                                                 const float* __restrict__ fcB,
                                                 const float* __restrict__ w,
                                                 float* __restrict__ out) {
    __shared__ float G[1024];
    __shared__ float S[32];
    const int tid = threadIdx.x;
    for (int e = tid; e < EPART; e += 256) {
        float acc = 0.0f;
        #pragma unroll 4
        for (int k = 0; k < NSLOT; ++k) acc += slotred[k * EPART + e];
        if (e < 1024) G[e] = acc; else S[e - 1024] = acc;
    }
    __syncthreads();

    if (tid < 32) {
        const int o = tid;
        // column-o softmax of w over i (numerically stable)
        float m = -INFINITY;
        for (int i = 0; i < 32; ++i) m = fmaxf(m, w[i * 32 + o]);
        float den = 0.0f;
        for (int i = 0; i < 32; ++i) den += expf(w[i * 32 + o] - m);
        const float bo = fcB[o];
        float acc = 0.0f;
        for (int i = 0; i < 32; ++i) {
            float t = bo * S[i];
            for (int j = 0; j < 32; ++j) t += G[i * 32 + j] * fcW[o * 32 + j];
            acc += (expf(w[i * 32 + o] - m) / den) * t;
        }
        out[o] = acc / (float)NSAMP;
    }
}

extern "C" void kernel_launch(void* const* d_in, const int* in_sizes, int n_in,
                              void* d_out, int out_size, void* d_ws, size_t ws_size,
                              hipStream_t stream) {
    (void)in_sizes; (void)n_in; (void)out_size; (void)ws_size;
    const float* x   = (const float*)d_in[0];
    const float* fea = (const float*)d_in[1];
    const float* fcW = (const float*)d_in[2];
    const float* fcB = (const float*)d_in[3];
    const float* w   = (const float*)d_in[4];
    float* out = (float*)d_out;

    float* partial = (float*)d_ws;                        // NBLK  * EPART
    float* slotred = partial + (size_t)NBLK * EPART;      // NSLOT * EPART
    // total ws use: (1024+32)*1056*4 B ~= 4.5 MB

    g_stage1<<<NBLK, 256, 0, stream>>>(x, fea, partial);
    const int n2a = NSLOT * EPART;
    g_stage2a<<<(n2a + 255) / 256, 256, 0, stream>>>(partial, slotred);
    g_stage2b<<<1, 256, 0, stream>>>(slotred, fcW, fcB, w, out);
}